// SpatioTemporalRoPE_17952963297532
// MI455X (gfx1250) — compile-verified
//
#include <hip/hip_runtime.h>
#include <hip/hip_bf16.h>

// SpatioTemporal RoPE for MI455X (gfx1250, wave32).
// Memory-bound streaming kernel (~805MB traffic -> ~35us @ 23.3TB/s).
// Temporal embedding GEMM (16x64 @ 64x96) done with V_WMMA_F32_16X16X4_F32.

typedef __attribute__((ext_vector_type(2))) float v2f;
typedef __attribute__((ext_vector_type(8))) float v8f;

#define BB 2
#define TT 8
#define NH 8
#define HH 64
#define WW 64
#define DD 96
#define NB 32          // DD/3 blocks
#define HWSZ (HH*WW)   // 4096
#define NROWS (BB*TT*NH*HH*WW)          // 524288 rows of 96 floats
#define QK_ELEMS ((size_t)NROWS * DD)   // 50331648 per tensor

// ---------------------------------------------------------------------------
// Kernel A: temp[16][96] = emb(16x64) @ Wt.T(64x96) + bt, via WMMA f32 16x16x4.
// emb[m][2j]   = cos(time_pos[m] * ft[j])
// emb[m][2j+1] = sin(time_pos[m] * ft[j])
// One block of 6 waves; wave w computes N-tile [16w, 16w+16).
// A-frag layout (16x4 f32): lane<16 -> M=lane, K=k0+{0,1}; lane>=16 -> K=k0+{2,3}
// B-frag layout (4x16 f32): vgpr r, lane L -> K = k0 + r + 2*(L>>4), N = L&15
// C/D layout (16x16 f32):   vgpr r, lane L -> M = r + 8*(L>>4),      N = L&15
// ---------------------------------------------------------------------------
__global__ void __launch_bounds__(192)
rope_temp_wmma_kernel(const float* __restrict__ time_pos,   // (B,T)=16
                      const float* __restrict__ freq_t,     // (32,)
                      const float* __restrict__ Wt,         // (96,64) row-major
                      const float* __restrict__ btv,        // (96,)
                      float* __restrict__ temp)             // (16,96) out
{
    const int lane  = threadIdx.x & 31;
    const int tile  = threadIdx.x >> 5;     // 0..5 -> N tile
    const int m     = lane & 15;            // A-frag row
    const int khalf = lane >> 4;            // 0: K+{0,1}, 1: K+{2,3}
    const int ng    = tile * 16 + (lane & 15);  // global output column (0..95)

    const float tp = time_pos[m];

#if __has_builtin(__builtin_amdgcn_wmma_f32_16x16x4_f32)
    v8f acc = {};
    #pragma unroll
    for (int k0 = 0; k0 < 64; k0 += 4) {
        const int kk = k0 + 2 * khalf;       // even K index this lane covers
        const int j  = kk >> 1;
        float s, c;
        __sincosf(tp * freq_t[j], &s, &c);
        v2f a; a.x = c; a.y = s;             // emb[m][kk], emb[m][kk+1]
        v2f b;
        b.x = Wt[(size_t)ng * 64 + kk];      // B[kk][ng]   = Wt[ng][kk]
        b.y = Wt[(size_t)ng * 64 + kk + 1];  // B[kk+1][ng]
        acc = __builtin_amdgcn_wmma_f32_16x16x4_f32(
            false, a, false, b, (short)0, acc, false, false);
    }
    const float bias = btv[ng];
    #pragma unroll
    for (int r = 0; r < 8; ++r) {
        const int M = r + 8 * khalf;
        temp[M * DD + ng] = acc[r] + bias;
    }
#else
    // Scalar fallback (should not be taken on gfx1250).
    if (threadIdx.x < DD) {
        const int n = threadIdx.x;
        for (int mm = 0; mm < 16; ++mm) {
            float accs = btv[n];
            const float t2 = time_pos[mm];
            for (int j = 0; j < 32; ++j) {
                float s, c;
                __sincosf(t2 * freq_t[j], &s, &c);
                accs += c * Wt[(size_t)n * 64 + 2 * j];
                accs += s * Wt[(size_t)n * 64 + 2 * j + 1];
            }
            temp[mm * DD + n] = accs;
        }
    }
#endif
}

// ---------------------------------------------------------------------------
// Kernel B: streaming rotate + add. One 32-lane wave per 96-float row;
// lane = 3-element block index nb. Coalesced b96 loads/stores, trig inline.
// ---------------------------------------------------------------------------
__global__ void __launch_bounds__(256)
rope_main_kernel(const float* __restrict__ q,
                 const float* __restrict__ k,
                 const float* __restrict__ spatial,   // (B,H,W,2) lat,lon
                 const float* __restrict__ freq_s,    // (NH,NB,2) fa,fb
                 const float* __restrict__ temp,      // (16,96) from kernel A
                 float* __restrict__ qo,
                 float* __restrict__ ko)
{
    const int lane  = threadIdx.x & 31;               // block index nb
    const int gwave = blockIdx.x * (blockDim.x >> 5) + (threadIdx.x >> 5);
    const int nwave = gridDim.x * (blockDim.x >> 5);

    for (int row = gwave; row < NROWS; row += nwave) {
        // row = (((b*T + t)*NH + head)*H + h)*W + w
        const int hw   = row & (HWSZ - 1);
        const int head = (row >> 12) & (NH - 1);
        const int bt   = row >> 15;                   // b*T + t  (0..15)
        const int b    = bt >> 3;

        // Broadcast-ish scalar-per-row loads (L2/WGP$ resident).
        const float* sp = spatial + ((size_t)(b * HWSZ + hw)) * 2;
        const float lat = sp[0];
        const float lon = sp[1];
        const float* fr = freq_s + ((size_t)(head * NB + lane)) * 2;
        const float fa  = fr[0];
        const float fb  = fr[1];

        float sA, cA, sB, cB;
        __sincosf(lon * fa, &sA, &cA);
        __sincosf(lat * fb, &sB, &cB);

        // Shared 3x3 rotation matrix for this (row, lane).
        const float r10 = -cB * sA, r11 = cB * cA, r12 = sB;
        const float r20 =  sB * sA, r21 = -sB * cA, r22 = cB;

        const size_t base = (size_t)row * DD + lane * 3;

        // Prefetch next grid-stride row of q/k (global_prefetch_b8).
        if (row + nwave < NROWS) {
            const size_t nbase = base + (size_t)nwave * DD;
            __builtin_prefetch(q + nbase, 0, 1);
            __builtin_prefetch(k + nbase, 0, 1);
        }

        const float3 xq = *(const float3*)(q + base);
        const float3 xk = *(const float3*)(k + base);
        const float3 tv = *(const float3*)(temp + (size_t)bt * DD + lane * 3);

        float3 oq, ok;
        oq.x = xq.x * cA  + xq.y * sA            + tv.x;
        oq.y = xq.x * r10 + xq.y * r11 + xq.z * r12 + tv.y;
        oq.z = xq.x * r20 + xq.y * r21 + xq.z * r22 + tv.z;
        ok.x = xk.x * cA  + xk.y * sA            + tv.x;
        ok.y = xk.x * r10 + xk.y * r11 + xk.z * r12 + tv.y;
        ok.z = xk.x * r20 + xk.y * r21 + xk.z * r22 + tv.z;

        *(float3*)(qo + base) = oq;
        *(float3*)(ko + base) = ok;
    }
}

extern "C" void kernel_launch(void* const* d_in, const int* in_sizes, int n_in,
                              void* d_out, int out_size, void* d_ws, size_t ws_size,
                              hipStream_t stream) {
    const float* q       = (const float*)d_in[0];
    const float* k       = (const float*)d_in[1];
    const float* spatial = (const float*)d_in[2];
    const float* tpos    = (const float*)d_in[3];
    const float* freq_s  = (const float*)d_in[4];
    const float* freq_t  = (const float*)d_in[5];
    const float* Wt      = (const float*)d_in[6];
    const float* btv     = (const float*)d_in[7];

    float* temp = (float*)d_ws;                 // 16*96 floats = 6KB scratch
    float* qo   = (float*)d_out;
    float* ko   = (float*)d_out + QK_ELEMS;

    // Kernel A: temporal embedding GEMM via WMMA (1 block, 6 waves).
    rope_temp_wmma_kernel<<<1, 192, 0, stream>>>(tpos, freq_t, Wt, btv, temp);

    // Kernel B: streaming rotate+add. 8 waves/block, 8 rows per wave.
    const int blocks = 8192;                    // 65536 waves, 524288 rows
    rope_main_kernel<<<blocks, 256, 0, stream>>>(q, k, spatial, freq_s, temp, qo, ko);
}